// TPAttention_51135880626981
// MI455X (gfx1250) — compile-verified
//
#include <hip/hip_runtime.h>
#include <hip/hip_bf16.h>
#include <cstdint>

#define B_  8
#define HID 2048
#define NH  32
#define NKV 8
#define D_  64
#define L_  8192
#define KP  8208          // padded score-row stride (8193 valid + pad)

typedef float v2f __attribute__((ext_vector_type(2)));
typedef float v8f __attribute__((ext_vector_type(8)));

#define WMMA_F32(a, b, c) \
  __builtin_amdgcn_wmma_f32_16x16x4_f32(false, (a), false, (b), (short)0, (c), false, false)

// ---- CDNA5 async global->LDS copy (ASYNCcnt-tracked), inline asm ----
template<int OFF>
__device__ __forceinline__ void async_b128(uint32_t lds, const float* src) {
  // INST_OFFSET is added to both the LDS and global addresses (ISA 10.7/15.18.3)
  asm volatile("global_load_async_to_lds_b128 %0, %1, off offset:%2"
               :: "v"(lds), "v"(src), "i"(OFF) : "memory");
}
__device__ __forceinline__ void async_half_row(uint32_t lds, const float* src) {
  async_b128<0>(lds, src);  async_b128<16>(lds, src);
  async_b128<32>(lds, src); async_b128<48>(lds, src);
  async_b128<64>(lds, src); async_b128<80>(lds, src);
  async_b128<96>(lds, src); async_b128<112>(lds, src);
}
__device__ __forceinline__ void wait_async8() { asm volatile("s_wait_asynccnt 8" ::: "memory"); }
__device__ __forceinline__ void wait_async0() { asm volatile("s_wait_asynccnt 0" ::: "memory"); }
__device__ __forceinline__ void wave_lds_fence() {
  __builtin_amdgcn_wave_barrier();
  asm volatile("" ::: "memory");
}
__device__ __forceinline__ uint32_t lds_addr(const void* p) {
  // generic shared-aperture address: low 32 bits are the LDS byte address
  return (uint32_t)(uintptr_t)p;
}

// ---------------- workspace layout (floats) ----------------
#define XPAD_OFF   ((size_t)0)          // 16 x 2048
#define ATTN_OFF   ((size_t)32768)      // 16 x 2048
#define QKVP_OFF   ((size_t)65536)      // 16 chunks x 16 x 3072
#define OPART_OFF  ((size_t)851968)     // 16 chunks x 16 x 2048
#define Q_OFF      ((size_t)1376256)    // B x NH x 64   (pre-scaled by 1/8)
#define KNEW_OFF   ((size_t)1392640)    // B x NKV x 64
#define VNEW_OFF   ((size_t)1396736)    // B x NKV x 64
#define RZ_OFF     ((size_t)1400832)    // B x NH
#define PVP_OFF    ((size_t)1401088)    // 8 chunks x B x NKV x 256
#define P_OFF      ((size_t)1532160)    // B x NH x KP

// ---------------- prep: zero-padded X ----------------
__global__ void k_prep_xpad(const float* __restrict__ x, float* __restrict__ xpad) {
  int idx = blockIdx.x * blockDim.x + threadIdx.x;
  if (idx >= 16 * HID) return;
  int r = idx >> 11, c = idx & (HID - 1);
  xpad[idx] = (r < B_) ? x[r * HID + c] : 0.f;
}

// ---------------- WMMA GEMM partial:  part[chunk] = A[16xK]chunk @ W[KxNw] --------
// stepped pointers: constant-stride address advance, no per-iter 64-bit multiplies
__global__ void k_gemm16_part(const float* __restrict__ A, const float* __restrict__ W,
                              float* __restrict__ part, int K, int Nw,
                              int part_stride, int coloff) {
  int lane  = threadIdx.x;
  int strip = blockIdx.x;
  int chunk = blockIdx.y;
  int kper  = K / gridDim.y;
  int k0    = chunk * kper;
  int row = lane & 15, kb = (lane >> 4) * 2, col = lane & 15, hi = lane >> 4;
  int cb = strip * 32;
  const float* aptr = A + (size_t)row * K + k0 + kb;
  const float* w0   = W + (size_t)(k0 + kb) * Nw + cb + col;
  const float* w1   = w0 + 16;
  const ptrdiff_t wrow  = Nw;            // +1 k-row
  const ptrdiff_t wstep = (ptrdiff_t)4 * Nw;
  v8f c0 = {}, c1 = {};
#pragma unroll 4
  for (int s = 0; s < kper; s += 4) {
    v2f a = *(const v2f*)aptr;
    v2f b0, b1;
    b0.x = w0[0]; b0.y = w0[wrow];
    b1.x = w1[0]; b1.y = w1[wrow];
    c0 = WMMA_F32(a, b0, c0);
    c1 = WMMA_F32(a, b1, c1);
    aptr += 4; w0 += wstep; w1 += wstep;
  }
  float* dst = part + (size_t)chunk * 16 * part_stride;
#pragma unroll
  for (int i = 0; i < 8; ++i) {
    int m = i + 8 * hi;
    dst[(size_t)m * part_stride + coloff + cb + col]      = c0[i];
    dst[(size_t)m * part_stride + coloff + cb + 16 + col] = c1[i];
  }
}

// ---------------- reduce QKV partials, apply RoPE, scale q by 1/8 ----------------
__global__ void k_rope_split(const float* __restrict__ part, const float* __restrict__ cosT,
                             const float* __restrict__ sinT, const int* __restrict__ lengths,
                             float* __restrict__ q, float* __restrict__ knew,
                             float* __restrict__ vnew) {
  int idx = blockIdx.x * blockDim.x + threadIdx.x;
  if (idx >= B_ * 3072) return;
  int b = idx / 3072, col = idx % 3072;
  auto rsum = [&](int c) -> float {
    float s = 0.f;
    for (int ch = 0; ch < 16; ++ch)
      s += part[(size_t)ch * 16 * 3072 + (size_t)b * 3072 + c];
    return s;
  };
  float v = rsum(col);
  int pos = lengths[0];
  if (col < 2048) {                       // Q
    int h = col >> 6, d = col & 63;
    float c = cosT[pos * 64 + d], s = sinT[pos * 64 + d];
    float partner = rsum((col & ~63) | ((d + 32) & 63));
    float rot = (d < 32) ? -partner : partner;
    q[((size_t)b * NH + h) * 64 + d] = (v * c + rot * s) * 0.125f;
  } else if (col < 2560) {                // K
    int cc = col - 2048, g = cc >> 6, d = cc & 63;
    float c = cosT[pos * 64 + d], s = sinT[pos * 64 + d];
    float partner = rsum(2048 + ((cc & ~63) | ((d + 32) & 63)));
    float rot = (d < 32) ? -partner : partner;
    knew[((size_t)b * NKV + g) * 64 + d] = v * c + rot * s;
  } else {                                // V
    int cc = col - 2560, g = cc >> 6, d = cc & 63;
    vnew[((size_t)b * NKV + g) * 64 + d] = v;
  }
}

// ---------------- scores: async double-buffered K tiles, barrier-free -------------
#define KSTR 68
__global__ void k_scores(const float* __restrict__ cache_k, const float* __restrict__ q,
                         const int* __restrict__ lengths, float* __restrict__ p) {
  __shared__ float klds[8][2][16 * KSTR];
  int b = blockIdx.z, g = blockIdx.y, chunk = blockIdx.x;
  int t = threadIdx.x, wave = t >> 5, lane = t & 31;
  int kmin = L_ - lengths[b];
  int row = lane & 15, kb = (lane >> 4) * 2, col = lane & 15, hi = lane >> 4;
  int base = chunk * 1024 + wave * 128;
  int t0 = (kmin > base) ? min(8, (kmin - base) >> 4) : 0;
  size_t prow_base = ((size_t)b * NH + g * 4) * KP;

  // fully-masked prefix tiles: store zero scores, no memory traffic
  if (col < 4) {
    float* prow = p + prow_base + (size_t)col * KP;
    for (int tile = 0; tile < t0; ++tile) {
      int key0 = base + tile * 16;
      for (int i = 0; i < 8; ++i) prow[key0 + i + 8 * hi] = 0.f;
    }
  }
  if (t0 >= 8) return;

  // Q^T operand lives in registers (reused by every tile)
  v2f qreg[16];
  if (col < 4) {
    const float* qp = q + ((size_t)b * NH + g * 4 + col) * 64 + kb;
#pragma unroll
    for (int s = 0; s < 16; ++s) qreg[s] = *(const v2f*)(qp + s * 4);
  } else {
#pragma unroll
    for (int s = 0; s < 16; ++s) { qreg[s].x = 0.f; qreg[s].y = 0.f; }
  }

  int lr = lane >> 1, half = lane & 1;
  auto issue = [&](int tt) {
    int key = base + tt * 16 + lr;
    const float* src = cache_k + (((size_t)b * L_ + key) * NKV + g) * D_ + half * 32;
    uint32_t dst = lds_addr(&klds[wave][tt & 1][lr * KSTR + half * 32]);
    async_half_row(dst, src);
  };
  issue(t0);
  for (int tile = t0; tile < 8; ++tile) {
    if (tile + 1 < 8) { issue(tile + 1); wait_async8(); }
    else              { wait_async0(); }
    int key0 = base + tile * 16;
    float* kt = &klds[wave][tile & 1][0];
    if (tile == t0 && key0 < kmin) {      // boundary tile: zero masked rows
      if (key0 + lr < kmin) {
        float4 z4 = make_float4(0.f, 0.f, 0.f, 0.f);
        float* dstp = kt + lr * KSTR + half * 32;
#pragma unroll
        for (int i = 0; i < 8; ++i) *(float4*)(dstp + i * 4) = z4;
      }
      wave_lds_fence();
    }
    v8f c = {};
    const float* ka = kt + row * KSTR + kb;
#pragma unroll
    for (int s = 0; s < 16; ++s) {
      v2f a = *(const v2f*)(ka + s * 4);     // immediate-offset ds_load_b64
      c = WMMA_F32(a, qreg[s], c);
    }
    if (col < 4) {
      float* prow = p + prow_base + (size_t)col * KP;
#pragma unroll
      for (int i = 0; i < 8; ++i) prow[key0 + i + 8 * hi] = c[i];
    }
  }
}

// ---------------- softmax per (b,h); masked slots keep exp(0-m) in Z but p=0 -------
__global__ void k_softmax(float* __restrict__ p, const float* __restrict__ q,
                          const float* __restrict__ knew, const int* __restrict__ lengths,
                          float* __restrict__ rZ) {
  __shared__ float red[256];
  int bh = blockIdx.x, b = bh >> 5, h = bh & 31, g = h >> 2;
  int t = threadIdx.x;
  const float* qv = q + ((size_t)b * NH + h) * 64;
  const float* kv = knew + ((size_t)b * NKV + g) * 64;
  float snew = 0.f;
  for (int d = 0; d < 64; ++d) snew += qv[d] * kv[d];   // q pre-scaled by 1/8
  float* prow = p + ((size_t)b * NH + h) * KP;
  int kmin = L_ - lengths[b];
  float m = -3.4e38f;
  for (int k = t; k < L_; k += 256) m = fmaxf(m, prow[k]);
  red[t] = m; __syncthreads();
  for (int o = 128; o > 0; o >>= 1) { if (t < o) red[t] = fmaxf(red[t], red[t + o]); __syncthreads(); }
  m = fmaxf(red[0], snew);
  __syncthreads();
  float z = 0.f;
  for (int k = t; k < L_; k += 256) {
    float e = __expf(prow[k] - m);
    z += e;                                  // masked slots stored 0 -> exp(-m) in Z
    prow[k] = (k >= kmin) ? e : 0.f;         // but p itself zeroed (V zeroed in ref)
  }
  red[t] = z; __syncthreads();
  for (int o = 128; o > 0; o >>= 1) { if (t < o) red[t] += red[t + o]; __syncthreads(); }
  float enew = __expf(snew - m);
  float Z = red[0] + enew;
  if (t == 0) { rZ[bh] = 1.f / Z; prow[L_] = enew; }
}

// ---------------- PV: async double-buffered V tiles, P direct from global ----------
#define VSTR 80
__global__ void k_pv(const float* __restrict__ cache_v, const float* __restrict__ p,
                     const float* __restrict__ rZ, const int* __restrict__ lengths,
                     float* __restrict__ pvpart) {
  __shared__ float vlds[8][2][16 * VSTR];
  __shared__ float outlds[8][256];
  int b = blockIdx.z, g = blockIdx.y, chunk = blockIdx.x;
  int t = threadIdx.x, wave = t >> 5, lane = t & 31;
  int kmin = L_ - lengths[b];
  int row = lane & 15, kb = (lane >> 4) * 2, col = lane & 15, hi = lane >> 4;
  int base = chunk * 1024 + wave * 128;
  int t0 = (kmin > base) ? min(8, (kmin - base) >> 4) : 0;
  v8f c0 = {}, c1 = {}, c2 = {}, c3 = {};

  if (t0 < 8) {
    int prow_h = (row < 4) ? row : 0;          // clamp: never form OOB pointer
    float rzv  = (row < 4) ? rZ[b * NH + g * 4 + row] : 0.f;
    const float* prow4 = p + ((size_t)b * NH + g * 4 + prow_h) * KP;
    int lr = lane >> 1, half = lane & 1;
    auto issue = [&](int tt) {
      int key = base + tt * 16 + lr;
      const float* src = cache_v + (((size_t)b * L_ + key) * NKV + g) * D_ + half * 32;
      uint32_t dst = lds_addr(&vlds[wave][tt & 1][lr * VSTR + half * 32]);
      async_half_row(dst, src);
    };
    issue(t0);
    for (int tile = t0; tile < 8; ++tile) {
      if (tile + 1 < 8) { issue(tile + 1); wait_async8(); }
      else              { wait_async0(); }
      int key0 = base + tile * 16;
      float* vt = &vlds[wave][tile & 1][0];
      // masked keys in boundary tile: p==0 there, so garbage V rows contribute 0
#pragma unroll
      for (int s = 0; s < 4; ++s) {
        v2f a; a.x = 0.f; a.y = 0.f;
        if (row < 4) {
          v2f pv = *(const v2f*)(prow4 + key0 + s * 4 + kb);
          a.x = pv.x * rzv; a.y = pv.y * rzv;
        }
        int kr0 = (s * 4 + kb) * VSTR, kr1 = kr0 + VSTR;
        v2f b0, b1, b2, b3;
        b0.x = vt[kr0 + col];      b0.y = vt[kr1 + col];
        b1.x = vt[kr0 + 16 + col]; b1.y = vt[kr1 + 16 + col];
        b2.x = vt[kr0 + 32 + col]; b2.y = vt[kr1 + 32 + col];
        b3.x = vt[kr0 + 48 + col]; b3.y = vt[kr1 + 48 + col];
        c0 = WMMA_F32(a, b0, c0);
        c1 = WMMA_F32(a, b1, c1);
        c2 = WMMA_F32(a, b2, c2);
        c3 = WMMA_F32(a, b3, c3);
      }
    }
  }
  if (hi == 0) {                                // valid heads: M = i (<4), lanes 0-15
#pragma unroll
    for (int i = 0; i < 4; ++i) {
      outlds[wave][i * 64 + col]      = c0[i];
      outlds[wave][i * 64 + 16 + col] = c1[i];
      outlds[wave][i * 64 + 32 + col] = c2[i];
      outlds[wave][i * 64 + 48 + col] = c3[i];
    }
  }
  __syncthreads();
  float s = 0.f;
#pragma unroll
  for (int w = 0; w < 8; ++w) s += outlds[w][t];
  pvpart[(((size_t)chunk * B_ + b) * NKV + g) * 256 + t] = s;
}

// ---------------- reduce PV partials + new-token term -> padded attn matrix --------
__global__ void k_pv_reduce(const float* __restrict__ pvpart, const float* __restrict__ p,
                            const float* __restrict__ rZ, const float* __restrict__ vnew,
                            float* __restrict__ attnpad) {
  int idx = blockIdx.x * blockDim.x + threadIdx.x;
  if (idx >= 16 * HID) return;
  int r = idx >> 11, c = idx & (HID - 1);
  if (r >= B_) { attnpad[idx] = 0.f; return; }
  int b = r, g = c >> 8, qh = (c >> 6) & 3, d = c & 63, h = g * 4 + qh;
  float s = 0.f;
  for (int ch = 0; ch < 8; ++ch)
    s += pvpart[(((size_t)ch * B_ + b) * NKV + g) * 256 + qh * 64 + d];
  float pn = p[((size_t)b * NH + h) * KP + L_] * rZ[b * NH + h];
  s += pn * vnew[((size_t)b * NKV + g) * 64 + d];
  attnpad[idx] = s;
}

// ---------------- reduce O partials -> final output --------------------------------
__global__ void k_o_reduce(const float* __restrict__ opart, float* __restrict__ out) {
  int idx = blockIdx.x * blockDim.x + threadIdx.x;
  if (idx >= B_ * HID) return;
  int b = idx >> 11, c = idx & (HID - 1);
  float s = 0.f;
  for (int ch = 0; ch < 16; ++ch)
    s += opart[(size_t)ch * 16 * HID + (size_t)b * HID + c];
  out[idx] = s;
}

extern "C" void kernel_launch(void* const* d_in, const int* in_sizes, int n_in,
                              void* d_out, int out_size, void* d_ws, size_t ws_size,
                              hipStream_t stream) {
  const float* x       = (const float*)d_in[0];
  const float* cosT    = (const float*)d_in[1];
  const float* sinT    = (const float*)d_in[2];
  const float* cache_k = (const float*)d_in[3];
  const float* cache_v = (const float*)d_in[4];
  const int*   lengths = (const int*)d_in[5];
  const float* Wq      = (const float*)d_in[6];
  const float* Wk      = (const float*)d_in[7];
  const float* Wv      = (const float*)d_in[8];
  const float* Wo      = (const float*)d_in[9];
  float* out = (float*)d_out;
  float* ws  = (float*)d_ws;

  float* xpad    = ws + XPAD_OFF;
  float* attnpad = ws + ATTN_OFF;
  float* qkvpart = ws + QKVP_OFF;
  float* opart   = ws + OPART_OFF;
  float* q       = ws + Q_OFF;
  float* knew    = ws + KNEW_OFF;
  float* vnew    = ws + VNEW_OFF;
  float* rZ      = ws + RZ_OFF;
  float* pvpart  = ws + PVP_OFF;
  float* p       = ws + P_OFF;

  k_prep_xpad<<<128, 256, 0, stream>>>(x, xpad);

  // QKV projections (K split 16 ways -> deterministic partial sums)
  k_gemm16_part<<<dim3(64, 16), 32, 0, stream>>>(xpad, Wq, qkvpart, HID, 2048, 3072, 0);
  k_gemm16_part<<<dim3(16, 16), 32, 0, stream>>>(xpad, Wk, qkvpart, HID, 512, 3072, 2048);
  k_gemm16_part<<<dim3(16, 16), 32, 0, stream>>>(xpad, Wv, qkvpart, HID, 512, 3072, 2560);

  k_rope_split<<<96, 256, 0, stream>>>(qkvpart, cosT, sinT, lengths, q, knew, vnew);

  k_scores<<<dim3(8, NKV, B_), 256, 0, stream>>>(cache_k, q, lengths, p);
  k_softmax<<<B_ * NH, 256, 0, stream>>>(p, q, knew, lengths, rZ);
  k_pv<<<dim3(8, NKV, B_), 256, 0, stream>>>(cache_v, p, rZ, lengths, pvpart);
  k_pv_reduce<<<128, 256, 0, stream>>>(pvpart, p, rZ, vnew, attnpad);

  // Output projection
  k_gemm16_part<<<dim3(64, 16), 32, 0, stream>>>(attnpad, Wo, opart, HID, 2048, 2048, 0);
  k_o_reduce<<<64, 256, 0, stream>>>(opart, out);
}